// DotLayer_70617852280973
// MI455X (gfx1250) — compile-verified
//
#include <hip/hip_runtime.h>

// out[r] = (a_r.b_r) * rsqrt(max(|a_r|^2,eps)) * rsqrt(max(|b_r|^2,eps))
// a,b: [65536, 256] fp32. Bandwidth-bound: 128 MiB read -> ~5.8us at 23.3 TB/s.
//
// MI455X plan: one wave per 16-row tile. K is staged in 64-float blocks from
// HBM into LDS with GLOBAL_LOAD_ASYNC_TO_LDS_B128 (512B fully-coalesced per
// instruction, no VGPR round trip, ASYNCcnt-tracked, double buffered), then
// consumed as WMMA 16x16x4 f32 operands via conflict-free ds_load_b64 reads.
// The 16x16x4 f32 A-operand and B-operand lane maps coincide, so the same two
// floats per lane feed either slot: three Gram tiles (a.bT, a.aT, b.bT)
// accumulate on the XDL pipe and we extract the diagonals.

typedef float v2f __attribute__((ext_vector_type(2)));
typedef float v8f __attribute__((ext_vector_type(8)));
typedef int   v4i __attribute__((vector_size(4 * sizeof(int))));

typedef __attribute__((address_space(1))) v4i* global_v4i_ptr;  // HIP "__device__"
typedef __attribute__((address_space(3))) v4i* shared_v4i_ptr;  // HIP "__shared__"

#define TILE_M          16
#define KC              64            // K floats staged per block
#define STRIDE          (KC + 4)      // pad -> conflict-free LDS banking
#define WAVES_PER_BLOCK 2
#define BLOCK_THREADS   (WAVES_PER_BLOCK * 32)

#if __has_builtin(__builtin_amdgcn_global_load_async_to_lds_b128)
#define HAVE_ASYNC_LDS 1
#else
#define HAVE_ASYNC_LDS 0
#endif

__device__ __forceinline__ void copy16_to_lds(const float* g, float* l) {
#if HAVE_ASYNC_LDS
    __builtin_amdgcn_global_load_async_to_lds_b128(
        (global_v4i_ptr)g, (shared_v4i_ptr)l, /*offset=*/0, /*cpol=*/0);
#else
    *(float4*)l = *(const float4*)g;   // sync fallback, still 512B/wave coalesced
#endif
}

template <int N>
__device__ __forceinline__ void wait_async() {
#if HAVE_ASYNC_LDS
#if __has_builtin(__builtin_amdgcn_s_wait_asynccnt)
    __builtin_amdgcn_s_wait_asynccnt((short)N);
#else
    asm volatile("s_wait_asynccnt %0" ::"n"(N));
#endif
#endif
    asm volatile("" ::: "memory");     // keep ds reads below the wait
}

__device__ __forceinline__ float vsel8(const v8f v, int r) {
    switch (r & 7) {
        case 0: return v[0];
        case 1: return v[1];
        case 2: return v[2];
        case 3: return v[3];
        case 4: return v[4];
        case 5: return v[5];
        case 6: return v[6];
        default: return v[7];
    }
}

__global__ __launch_bounds__(BLOCK_THREADS)
void cosine_diag_wmma_kernel(const float* __restrict__ a,
                             const float* __restrict__ b,
                             float* __restrict__ out,
                             int ntiles, int K)
{
    // Per-wave private staging buffers (double buffered): ~34.8 KB total.
    __shared__ float sA[WAVES_PER_BLOCK][2][TILE_M * STRIDE];
    __shared__ float sB[WAVES_PER_BLOCK][2][TILE_M * STRIDE];

    const int lane = threadIdx.x & 31;
    const int wave = threadIdx.x >> 5;
    const int tile = blockIdx.x * WAVES_PER_BLOCK + wave;
    if (tile >= ntiles) return;        // wave-uniform: EXEC all-ones at WMMAs

    const int m    = lane & 15;        // M/N index this lane serves in WMMA
    const int h    = lane >> 4;        // K-pair half: 0 -> k+{0,1}, 1 -> k+{2,3}
    const int row0 = tile * TILE_M;

    // Staging role: lane covers (row = 2i+h, 16B segment s=m) -> two contiguous
    // 256B row-chunks per instruction (fully coalesced 512B).
    const int seg  = m;
    const int half = h;

    // ---- stage block 0 ----
    {
#pragma unroll
        for (int i = 0; i < 8; ++i) {
            const int row = 2 * i + half;
            const size_t goff = (size_t)(row0 + row) * (size_t)K + (size_t)(4 * seg);
            copy16_to_lds(a + goff, &sA[wave][0][row * STRIDE + 4 * seg]);
            copy16_to_lds(b + goff, &sB[wave][0][row * STRIDE + 4 * seg]);
        }
    }

    v8f z = {0.f, 0.f, 0.f, 0.f, 0.f, 0.f, 0.f, 0.f};
    v8f ab0 = z, ab1 = z, aa0 = z, aa1 = z, bb0 = z, bb1 = z;

    const int KB = K / KC;             // 4 blocks of 64 floats
    for (int blk = 0; blk < KB; ++blk) {
        const int cur = blk & 1;

        if (blk + 1 < KB) {
            // Prefetch next block into the other buffer, then wait for current.
            const int nxt = cur ^ 1;
            const int kb  = (blk + 1) * KC;
#pragma unroll
            for (int i = 0; i < 8; ++i) {
                const int row = 2 * i + half;
                const size_t goff =
                    (size_t)(row0 + row) * (size_t)K + (size_t)(kb + 4 * seg);
                copy16_to_lds(a + goff, &sA[wave][nxt][row * STRIDE + 4 * seg]);
                copy16_to_lds(b + goff, &sB[wave][nxt][row * STRIDE + 4 * seg]);
            }
            wait_async<16>();          // in-order ASYNCcnt: block `blk` resident
        } else {
            wait_async<0>();
        }

        const float* la = &sA[wave][cur][m * STRIDE + 2 * h];
        const float* lb = &sB[wave][cur][m * STRIDE + 2 * h];

#pragma unroll
        for (int c = 0; c < KC; c += 8) {
            // ds_load_b64: all 32 lanes hit distinct banks (STRIDE padding).
            v2f av0 = *(const v2f*)(la + c);
            v2f bv0 = *(const v2f*)(lb + c);
            v2f av1 = *(const v2f*)(la + c + 4);
            v2f bv1 = *(const v2f*)(lb + c + 4);
            // Two independent accumulator sets shorten the WMMA RAW chains.
            ab0 = __builtin_amdgcn_wmma_f32_16x16x4_f32(false, av0, false, bv0,
                                                        (short)0, ab0, false, false);
            aa0 = __builtin_amdgcn_wmma_f32_16x16x4_f32(false, av0, false, av0,
                                                        (short)0, aa0, false, false);
            bb0 = __builtin_amdgcn_wmma_f32_16x16x4_f32(false, bv0, false, bv0,
                                                        (short)0, bb0, false, false);
            ab1 = __builtin_amdgcn_wmma_f32_16x16x4_f32(false, av1, false, bv1,
                                                        (short)0, ab1, false, false);
            aa1 = __builtin_amdgcn_wmma_f32_16x16x4_f32(false, av1, false, av1,
                                                        (short)0, aa1, false, false);
            bb1 = __builtin_amdgcn_wmma_f32_16x16x4_f32(false, bv1, false, bv1,
                                                        (short)0, bb1, false, false);
        }
    }

    const v8f abv = ab0 + ab1;
    const v8f aav = aa0 + aa1;
    const v8f bbv = bb0 + bb1;

    // C layout: VGPR r, lane L holds C[M = r + 8*(L>>4)][N = L&15].
    // Diagonal: lanes 0-7 -> VGPR m, lanes 24-31 -> VGPR m-8.
    const bool active = (h == 0) ? (m < 8) : (m >= 8);
    const int  r      = (m - 8 * h) & 7;

    const float ab = vsel8(abv, r);
    const float aa = vsel8(aav, r);
    const float bb = vsel8(bbv, r);

    if (active) {
        const float eps = 1e-12f;
        const float inv = rsqrtf(fmaxf(aa, eps)) * rsqrtf(fmaxf(bb, eps));
        out[row0 + m] = ab * inv;
    }
}

extern "C" void kernel_launch(void* const* d_in, const int* in_sizes, int n_in,
                              void* d_out, int out_size, void* d_ws, size_t ws_size,
                              hipStream_t stream) {
    const float* a = (const float*)d_in[0];
    const float* b = (const float*)d_in[1];
    float* out = (float*)d_out;

    const int rows   = out_size;            // B*N = 65536
    const int K      = in_sizes[0] / rows;  // 256
    const int ntiles = rows / TILE_M;       // 4096
    const int blocks = (ntiles + WAVES_PER_BLOCK - 1) / WAVES_PER_BLOCK;

    cosine_diag_wmma_kernel<<<blocks, BLOCK_THREADS, 0, stream>>>(a, b, out, ntiles, K);
}